// GNNModel_79568564126007
// MI455X (gfx1250) — compile-verified
//
#include <hip/hip_runtime.h>

typedef __attribute__((ext_vector_type(2))) float v2f;
typedef __attribute__((ext_vector_type(8))) float v8f;

#define N_NODES 50000
#define N_EDGES 800000
#define N_GRAPH 512
#define F_IN    64
#define HDIM    128
#define NLAYERS 3
#define OUTD    12
#define LN_EPS  1e-5f

// ---------------- degree / GCN norm ----------------
__global__ void k_init_deg(float* deg) {
    int i = blockIdx.x * blockDim.x + threadIdx.x;
    if (i < N_NODES) deg[i] = 1.0f;             // self-loop contributes 1
}

__global__ void k_deg_scatter(const int* __restrict__ ei, float* deg) {
    int e = blockIdx.x * blockDim.x + threadIdx.x;
    if (e < N_EDGES) atomicAdd(&deg[ei[N_EDGES + e]], 1.0f);   // dst row
}

__global__ void k_dinv(float* deg) {
    int i = blockIdx.x * blockDim.x + threadIdx.x;
    if (i < N_NODES) { float d = deg[i]; deg[i] = (d > 0.0f) ? rsqrtf(d) : 0.0f; }
}

// ---------------- WMMA fp32 GEMM: C[M,NCOLS] = A[M,K] @ B[K,NCOLS] (+bias, +relu) ---------
// blockDim = 32 (one wave). grid = (M/16, ceil(NCOLS/(16*NT))). M multiple of 16.
// Each wave computes 16 rows x (16*NT) cols. K and NCOLS are compile-time so:
//   * the k-loop fully unrolls and is software-pipelined one step ahead (double-buffered
//     A/B operands: loads for step k+4 in flight while WMMAs of step k execute), and
//   * all strides fold into the 24-bit immediate offsets of global_load/store — the
//     per-lane base addresses (A row, B column, C column) are computed exactly once.
//
// V_WMMA_F32_16X16X4_F32 layouts (ISA 7.12.2):
//   A (16x4): lanes 0-15 rows 0-15 hold K={k,k+1}; lanes 16-31 same rows hold K={k+2,k+3}
//   B (4x16): lanes 0-15 cols hold rows {k,k+1}; lanes 16-31 hold rows {k+2,k+3}
//   C (16x16): VGPR v -> row v (lanes 0-15) / row v+8 (lanes 16-31)
//
// Out-of-range columns: lane n's B values only feed C column n (same lane), so we clamp the
// load address (branch-free, keeps EXEC full for WMMA) and drop the result at the store.
template <int K, int NT, int NCOLS>
__global__ void k_wmma_gemm(const float* __restrict__ A, const float* __restrict__ B,
                            const float* __restrict__ bias, float* __restrict__ C,
                            int M, int do_relu) {
    const int lane = threadIdx.x;
    const int half = lane >> 4;            // 0 -> K+{0,1}, 1 -> K+{2,3}
    const int l16  = lane & 15;
    const int row  = blockIdx.x * 16 + l16;    // A row for this lane (always < M)
    const int ko   = 2 * half;                 // lane's K sub-offset within a 4-step

    int col[NT];
    const float* Bcol[NT];                      // per-lane B column base (computed once)
    #pragma unroll
    for (int t = 0; t < NT; ++t) {
        col[t] = blockIdx.y * (16 * NT) + t * 16 + l16;
        Bcol[t] = B + min(col[t], NCOLS - 1);   // clamped, branch-free load base
    }

    const float* Arow = A + (size_t)row * K + ko;   // per-lane A base (computed once)
    v8f acc[NT];
    #pragma unroll
    for (int t = 0; t < NT; ++t) acc[t] = (v8f){};

    // ---- pipeline prologue: operands for k = 0 ----
    v2f a_cur, b_cur[NT];
    a_cur.x = Arow[0];
    a_cur.y = Arow[1];
    #pragma unroll
    for (int t = 0; t < NT; ++t) {
        b_cur[t].x = Bcol[t][(ko + 0) * NCOLS];
        b_cur[t].y = Bcol[t][(ko + 1) * NCOLS];
    }

    #pragma unroll
    for (int k = 0; k < K; k += 4) {
        v2f a_nxt = a_cur, b_nxt[NT];
        #pragma unroll
        for (int t = 0; t < NT; ++t) b_nxt[t] = b_cur[t];
        if (k + 4 < K) {                       // compile-time under full unroll
            a_nxt.x = Arow[k + 4];
            a_nxt.y = Arow[k + 5];
            #pragma unroll
            for (int t = 0; t < NT; ++t) {
                b_nxt[t].x = Bcol[t][(k + 4 + ko) * NCOLS];
                b_nxt[t].y = Bcol[t][(k + 5 + ko) * NCOLS];
            }
        }
        #pragma unroll
        for (int t = 0; t < NT; ++t) {
            acc[t] = __builtin_amdgcn_wmma_f32_16x16x4_f32(
                /*neg_a=*/false, a_cur, /*neg_b=*/false, b_cur[t],
                /*c_mod=*/(short)0, acc[t], /*reuse_a=*/false, /*reuse_b=*/false);
        }
        a_cur = a_nxt;
        #pragma unroll
        for (int t = 0; t < NT; ++t) b_cur[t] = b_nxt[t];
    }

    const int rbase = blockIdx.x * 16 + 8 * half;
    #pragma unroll
    for (int t = 0; t < NT; ++t) {
        if (col[t] < NCOLS) {
            const float bv = bias ? bias[col[t]] : 0.0f;
            float* Ccol = C + (size_t)rbase * NCOLS + col[t];
            #pragma unroll
            for (int v = 0; v < 8; ++v) {
                float r = acc[t][v] + bv;
                if (do_relu) r = fmaxf(r, 0.0f);
                Ccol[v * NCOLS] = r;
            }
        }
    }
}

// ---------------- LayerNorm (+opt bias add, +relu, +opt residual), one wave per row -------
__global__ void k_ln_relu(const float* __restrict__ t, const float* __restrict__ addb,
                          const float* __restrict__ gamma, const float* __restrict__ beta,
                          const float* __restrict__ resid, float* __restrict__ out, int M) {
    int g = blockIdx.x * blockDim.x + threadIdx.x;
    int row = g >> 5;
    int lane = g & 31;
    if (row >= M) return;
    int c = lane * 4;
    float4 v = *(const float4*)(t + (size_t)row * HDIM + c);
    if (addb) {
        float4 b = *(const float4*)(addb + c);
        v.x += b.x; v.y += b.y; v.z += b.z; v.w += b.w;
    }
    float s  = v.x + v.y + v.z + v.w;
    float sq = v.x * v.x + v.y * v.y + v.z * v.z + v.w * v.w;
    #pragma unroll
    for (int m = 16; m >= 1; m >>= 1) {
        s  += __shfl_xor(s,  m, 32);
        sq += __shfl_xor(sq, m, 32);
    }
    const float mean = s * (1.0f / HDIM);
    const float var  = sq * (1.0f / HDIM) - mean * mean;
    const float rstd = rsqrtf(var + LN_EPS);
    float4 gm = *(const float4*)(gamma + c);
    float4 bt = *(const float4*)(beta  + c);
    float4 o;
    o.x = fmaxf((v.x - mean) * rstd * gm.x + bt.x, 0.0f);
    o.y = fmaxf((v.y - mean) * rstd * gm.y + bt.y, 0.0f);
    o.z = fmaxf((v.z - mean) * rstd * gm.z + bt.z, 0.0f);
    o.w = fmaxf((v.w - mean) * rstd * gm.w + bt.w, 0.0f);
    if (resid) {
        float4 r = *(const float4*)(resid + (size_t)row * HDIM + c);
        o.x += r.x; o.y += r.y; o.z += r.z; o.w += r.w;
    }
    *(float4*)(out + (size_t)row * HDIM + c) = o;
}

// ---------------- self-loop term initializes agg (no atomics needed, runs alone) ----------
__global__ void k_selfloop(const float* __restrict__ m, const float* __restrict__ dinv,
                           float* __restrict__ agg) {
    int g = blockIdx.x * blockDim.x + threadIdx.x;           // over N*32 float4 chunks
    if (g >= N_NODES * 32) return;
    int i = g >> 5, c = (g & 31) * 4;
    float w = dinv[i] * dinv[i];
    float4 v = *(const float4*)(m + (size_t)i * HDIM + c);
    float4 o = { v.x * w, v.y * w, v.z * w, v.w * w };
    *(float4*)(agg + (size_t)i * HDIM + c) = o;
}

// ---------------- edge gather + scatter-add (L2-resident atomics) -------------------------
__global__ void k_edge_scatter(const int* __restrict__ ei, const float* __restrict__ m,
                               const float* __restrict__ dinv, float* __restrict__ agg) {
    int g = blockIdx.x * blockDim.x + threadIdx.x;           // over E*32 float4 chunks
    int e = g >> 5;
    if (e >= N_EDGES) return;
    int c = (g & 31) * 4;
    int s = ei[e];
    int d = ei[N_EDGES + e];
    float w = dinv[s] * dinv[d];
    float4 v = *(const float4*)(m + (size_t)s * HDIM + c);
    float* a = agg + (size_t)d * HDIM + c;
    atomicAdd(a + 0, v.x * w);
    atomicAdd(a + 1, v.y * w);
    atomicAdd(a + 2, v.z * w);
    atomicAdd(a + 3, v.w * w);
}

// ---------------- pooling ----------------
__global__ void k_pool_zero(float* pooled, float* counts) {
    int g = blockIdx.x * blockDim.x + threadIdx.x;
    if (g < N_GRAPH * HDIM) pooled[g] = 0.0f;
    if (g < N_GRAPH) counts[g] = 0.0f;
}

__global__ void k_pool_count(const int* __restrict__ bids, float* counts) {
    int i = blockIdx.x * blockDim.x + threadIdx.x;
    if (i < N_NODES) atomicAdd(&counts[bids[i]], 1.0f);
}

__global__ void k_pool_sum(const int* __restrict__ bids, const float* __restrict__ h,
                           float* __restrict__ pooled) {
    int g = blockIdx.x * blockDim.x + threadIdx.x;
    if (g >= N_NODES * 32) return;
    int i = g >> 5, c = (g & 31) * 4;
    int b = bids[i];
    float4 v = *(const float4*)(h + (size_t)i * HDIM + c);
    float* p = pooled + (size_t)b * HDIM + c;
    atomicAdd(p + 0, v.x);
    atomicAdd(p + 1, v.y);
    atomicAdd(p + 2, v.z);
    atomicAdd(p + 3, v.w);
}

__global__ void k_pool_div(float* pooled, const float* __restrict__ counts) {
    int g = blockIdx.x * blockDim.x + threadIdx.x;
    if (g < N_GRAPH * HDIM) pooled[g] /= fmaxf(counts[g >> 7], 1.0f);   // HDIM == 128
}

// ---------------- launch ----------------
extern "C" void kernel_launch(void* const* d_in, const int* in_sizes, int n_in,
                              void* d_out, int out_size, void* d_ws, size_t ws_size,
                              hipStream_t stream) {
    const float* x        = (const float*)d_in[0];
    const int*   ei       = (const int*)  d_in[1];   // [2,E]: row0=src, row1=dst
    const int*   bids     = (const int*)  d_in[2];
    const float* W_embed  = (const float*)d_in[3];
    const float* b_embed  = (const float*)d_in[4];
    const float* g_embed  = (const float*)d_in[5];
    const float* be_embed = (const float*)d_in[6];
    const float* W_conv   = (const float*)d_in[7];   // [L,H,H]
    const float* b_conv   = (const float*)d_in[8];   // [L,H]
    const float* g_ln     = (const float*)d_in[9];   // [L,H]
    const float* b_ln     = (const float*)d_in[10];  // [L,H]
    const float* W_r1     = (const float*)d_in[11];  // [H,H/2]
    const float* b_r1     = (const float*)d_in[12];
    const float* W_r2     = (const float*)d_in[13];  // [H/2,OUT]
    const float* b_r2     = (const float*)d_in[14];
    float* out = (float*)d_out;                      // [G,OUT]

    float* ws     = (float*)d_ws;
    float* h      = ws;                                       // N*H
    float* m      = h + (size_t)N_NODES * HDIM;               // N*H
    float* agg    = m + (size_t)N_NODES * HDIM;               // N*H
    float* dinv   = agg + (size_t)N_NODES * HDIM;             // N
    float* pooled = dinv + N_NODES;                           // G*H
    float* counts = pooled + (size_t)N_GRAPH * HDIM;          // G
    float* r1     = counts + N_GRAPH;                         // G*(H/2)

    const int TB = 256;
    const int gN   = (N_NODES + TB - 1) / TB;
    const int gE   = (N_EDGES + TB - 1) / TB;
    const int gNH4 = (N_NODES * 32 + TB - 1) / TB;
    const int gEH4 = (N_EDGES * 32 + TB - 1) / TB;
    const int gGH  = (N_GRAPH * HDIM + TB - 1) / TB;

    // GCN normalization: deg (with self-loops) -> d^{-1/2}
    k_init_deg   <<<gN, TB, 0, stream>>>(dinv);
    k_deg_scatter<<<gE, TB, 0, stream>>>(ei, dinv);
    k_dinv       <<<gN, TB, 0, stream>>>(dinv);

    // embedding: h = relu(LN(x @ W_embed + b_embed))
    k_wmma_gemm<F_IN, 2, HDIM><<<dim3(N_NODES / 16, HDIM / 32), 32, 0, stream>>>(
        x, W_embed, b_embed, m, N_NODES, 0);
    k_ln_relu<<<gNH4, TB, 0, stream>>>(m, nullptr, g_embed, be_embed, nullptr, h, N_NODES);

    // GCN layers
    for (int l = 0; l < NLAYERS; ++l) {
        k_wmma_gemm<HDIM, 2, HDIM><<<dim3(N_NODES / 16, HDIM / 32), 32, 0, stream>>>(
            h, W_conv + (size_t)l * HDIM * HDIM, nullptr, m, N_NODES, 0);
        k_selfloop    <<<gNH4, TB, 0, stream>>>(m, dinv, agg);
        k_edge_scatter<<<gEH4, TB, 0, stream>>>(ei, m, dinv, agg);
        // h = relu(LN(agg + b_conv)) + h   (in place: residual read before write per thread)
        k_ln_relu<<<gNH4, TB, 0, stream>>>(agg, b_conv + (size_t)l * HDIM,
                                           g_ln + (size_t)l * HDIM, b_ln + (size_t)l * HDIM,
                                           h, h, N_NODES);
    }

    // global mean pooling
    k_pool_zero <<<gGH, TB, 0, stream>>>(pooled, counts);
    k_pool_count<<<gN, TB, 0, stream>>>(bids, counts);
    k_pool_sum  <<<gNH4, TB, 0, stream>>>(bids, h, pooled);
    k_pool_div  <<<gGH, TB, 0, stream>>>(pooled, counts);

    // readout MLP: relu(pooled @ W_r1 + b_r1) @ W_r2 + b_r2
    k_wmma_gemm<HDIM, 2, HDIM / 2><<<dim3(N_GRAPH / 16, (HDIM / 2) / 32), 32, 0, stream>>>(
        pooled, W_r1, b_r1, r1, N_GRAPH, 1);
    k_wmma_gemm<HDIM / 2, 1, OUTD><<<dim3(N_GRAPH / 16, 1), 32, 0, stream>>>(
        r1, W_r2, b_r2, out, N_GRAPH, 0);
}